// PointerNet_42502996361356
// MI455X (gfx1250) — compile-verified
//
#include <hip/hip_runtime.h>
#include <math.h>

#define INP   512
#define HID   512
#define NSEQ  256
#define BATCH 64
#define GATESZ 2048  // 4*HID

typedef __attribute__((ext_vector_type(2))) float v2f;
typedef __attribute__((ext_vector_type(8))) float v8f;

// ---------------------------------------------------------------------------
// WMMA helper (fp32 16x16x4 — full precision keeps argmax tie-breaks stable)
// ---------------------------------------------------------------------------
static __device__ __forceinline__ v8f wmma4(v2f a, v2f b, v8f c) {
    return __builtin_amdgcn_wmma_f32_16x16x4_f32(false, a, false, b, (short)0, c,
                                                 false, false);
}

// A fragment (16x4, MxK) or B fragment (4x16, from W[n][k] row-major):
// lane<16 holds (row=base+lane, k0..k0+1); lane>=16 holds k0+2..k0+3.
static __device__ __forceinline__ v2f frag_ld(const float* __restrict__ p, int ld,
                                              int r0, int k0, int lane) {
    int row = r0 + (lane & 15);
    int kk  = k0 + ((lane >> 4) << 1);
    return *(const v2f*)(p + (size_t)row * ld + kk);
}

// ---------------------------------------------------------------------------
// Grid-wide barrier (persistent LSTM kernel). bar[0]=count, bar[1]=generation.
// ---------------------------------------------------------------------------
static __device__ __forceinline__ void grid_barrier(unsigned* bar, unsigned nblocks) {
    __threadfence();
    __syncthreads();
    if (threadIdx.x == 0) {
        unsigned gen = __hip_atomic_load(bar + 1, __ATOMIC_ACQUIRE,
                                         __HIP_MEMORY_SCOPE_AGENT);
        unsigned arrived = __hip_atomic_fetch_add(bar, 1u, __ATOMIC_ACQ_REL,
                                                  __HIP_MEMORY_SCOPE_AGENT);
        if (arrived == nblocks - 1) {
            __hip_atomic_store(bar, 0u, __ATOMIC_RELEASE, __HIP_MEMORY_SCOPE_AGENT);
            __hip_atomic_fetch_add(bar + 1, 1u, __ATOMIC_ACQ_REL,
                                   __HIP_MEMORY_SCOPE_AGENT);
        } else {
            while (__hip_atomic_load(bar + 1, __ATOMIC_ACQUIRE,
                                     __HIP_MEMORY_SCOPE_AGENT) == gen) {
                __builtin_amdgcn_s_sleep(1);
            }
        }
    }
    __syncthreads();
}

// ---------------------------------------------------------------------------
// Init: zero barrier state and h0/c0 (encoder initial state).
// ---------------------------------------------------------------------------
__global__ void init_kernel(unsigned* bar, float* h0, float* c) {
    if (threadIdx.x == 0) { bar[0] = 0u; bar[1] = 0u; }
    for (int i = threadIdx.x; i < BATCH * HID; i += blockDim.x) {
        h0[i] = 0.f;
        c[i]  = 0.f;
    }
}

// ---------------------------------------------------------------------------
// Input projection:  xprojT[t][b][g] = sum_k X[b,t,k]*Wih[g,k] + bih[g]+bhh[g]
// M = B*N = 16384, K = 512, Ngates = 2048.  One wave computes a 16x64 tile.
// ---------------------------------------------------------------------------
__global__ void __launch_bounds__(256) xproj_kernel(
    const float* __restrict__ X,    // [B, N, INP]
    const float* __restrict__ Wih,  // [GATESZ, INP]
    const float* __restrict__ bih,
    const float* __restrict__ bhh,
    float* __restrict__ xprojT)     // [N, B, GATESZ]
{
    const int lane = threadIdx.x & 31;
    const int wid  = (blockIdx.x * blockDim.x + threadIdx.x) >> 5;
    const int ntile = wid & 31;   // 32 tiles of 64 gate columns
    const int mtile = wid >> 5;   // 1024 tiles of 16 rows
    if (mtile >= (BATCH * NSEQ) / 16) return;
    const int m0 = mtile * 16;
    const int n0 = ntile * 64;

    v8f acc0 = {}, acc1 = {}, acc2 = {}, acc3 = {};
#pragma unroll 2
    for (int k0 = 0; k0 < INP; k0 += 4) {
        v2f a  = frag_ld(X,   INP, m0,       k0, lane);
        v2f b0 = frag_ld(Wih, INP, n0 +  0,  k0, lane);
        v2f b1 = frag_ld(Wih, INP, n0 + 16,  k0, lane);
        v2f b2 = frag_ld(Wih, INP, n0 + 32,  k0, lane);
        v2f b3 = frag_ld(Wih, INP, n0 + 48,  k0, lane);
        __builtin_prefetch(Wih + (size_t)(n0 + (lane & 15)) * INP + k0 + 16, 0, 3);
        acc0 = wmma4(a, b0, acc0);
        acc1 = wmma4(a, b1, acc1);
        acc2 = wmma4(a, b2, acc2);
        acc3 = wmma4(a, b3, acc3);
    }

    const int b_idx = m0 >> 8;           // batch is constant inside a 16-row tile
    v8f accs[4] = {acc0, acc1, acc2, acc3};
#pragma unroll
    for (int sub = 0; sub < 4; ++sub) {
        const int col  = n0 + sub * 16 + (lane & 15);
        const float bs = bih[col] + bhh[col];
#pragma unroll
        for (int v = 0; v < 8; ++v) {
            const int m = m0 + v + ((lane >> 4) << 3);
            const int t = m & (NSEQ - 1);
            xprojT[((size_t)t * BATCH + b_idx) * GATESZ + col] = accs[sub][v] + bs;
        }
    }
}

// ---------------------------------------------------------------------------
// Persistent fused LSTM: 16 blocks x 256 threads = 128 waves.
// Wave (mtile, strip) owns batches m0..m0+15 and hidden cols n0..n0+15 for
// ALL FOUR gates (Whh rows g*HID+n0).  The i/f/g/o values for a given (b,j)
// land in the same (VGPR,lane) slot of the 4 accumulators, so the cell update
// fuses locally: no gates scratch, ONE grid barrier per step.
// h is double-buffered (read h[t&1], write h[(t+1)&1]); c is wave-private.
// ---------------------------------------------------------------------------
__global__ void __launch_bounds__(256) lstm_kernel(
    const float* __restrict__ xprojT,  // [N, B, GATESZ] (bias folded in)
    const float* __restrict__ Whh,     // [GATESZ, HID]
    const int*   __restrict__ lens,    // [B]
    float* __restrict__ hbuf0,         // [B, HID] ping
    float* __restrict__ hbuf1,         // [B, HID] pong
    float* __restrict__ c,             // [B, HID]
    float* __restrict__ out,           // [B, N, HID]
    unsigned* bar)
{
    const unsigned nb = gridDim.x;
    const int lane  = threadIdx.x & 31;
    const int wid   = blockIdx.x * (blockDim.x >> 5) + (threadIdx.x >> 5); // 0..127
    const int mtile = wid & 3;    // 4 tiles of 16 batch rows
    const int strip = wid >> 2;   // 32 strips of 16 hidden cols
    const int m0 = mtile * 16;
    const int n0 = strip * 16;
    const int j  = n0 + (lane & 15);   // hidden column owned by this lane

    for (int t = 0; t < NSEQ; ++t) {
        const float* hr = (t & 1) ? hbuf1 : hbuf0;
        float*       hw = (t & 1) ? hbuf0 : hbuf1;

        // ---- GEMM: acc[g] = h @ Whh[g*HID+n0 .. +15, :]^T ----
        v8f acc[4] = {v8f{}, v8f{}, v8f{}, v8f{}};
#pragma unroll 2
        for (int k0 = 0; k0 < HID; k0 += 4) {
            v2f a  = frag_ld(hr,  HID, m0,           k0, lane);
            v2f b0 = frag_ld(Whh, HID, n0 + 0 * HID, k0, lane);
            v2f b1 = frag_ld(Whh, HID, n0 + 1 * HID, k0, lane);
            v2f b2 = frag_ld(Whh, HID, n0 + 2 * HID, k0, lane);
            v2f b3 = frag_ld(Whh, HID, n0 + 3 * HID, k0, lane);
            acc[0] = wmma4(a, b0, acc[0]);
            acc[1] = wmma4(a, b1, acc[1]);
            acc[2] = wmma4(a, b2, acc[2]);
            acc[3] = wmma4(a, b3, acc[3]);
        }

        // ---- fused cell update for the 8 (b, j) elements of this lane ----
        const float* xp = xprojT + (size_t)t * BATCH * GATESZ;
#pragma unroll
        for (int v = 0; v < 8; ++v) {
            const int m   = m0 + v + ((lane >> 4) << 3);   // batch row
            const size_t gbase = (size_t)m * GATESZ + j;
            const float gi = acc[0][v] + xp[gbase + 0 * HID];
            const float gf = acc[1][v] + xp[gbase + 1 * HID];
            const float gg = acc[2][v] + xp[gbase + 2 * HID];
            const float go = acc[3][v] + xp[gbase + 3 * HID];
            const float si = 1.f / (1.f + expf(-gi));
            const float sf = 1.f / (1.f + expf(-gf));
            const float so = 1.f / (1.f + expf(-go));
            const float tg = tanhf(gg);
            const size_t idx = (size_t)m * HID + j;
            const float cc = c[idx];
            const float hh = hr[idx];
            const float cn = sf * cc + si * tg;
            const float hn = so * tanhf(cn);
            const bool valid = t < lens[m];
            hw[idx] = valid ? hn : hh;
            c[idx]  = valid ? cn : cc;
            out[((size_t)m * NSEQ + t) * HID + j] = valid ? hn : 0.f;
        }
        grid_barrier(bar, nb);
    }
}

// ---------------------------------------------------------------------------
// u1 = W1^T v, u2 = W2^T v   (each 512-long; columns coalesce across threads)
// ---------------------------------------------------------------------------
__global__ void uvec_kernel(const float* __restrict__ W1, const float* __restrict__ W2,
                            const float* __restrict__ v,
                            float* __restrict__ u1, float* __restrict__ u2) {
    const int g = blockIdx.x * blockDim.x + threadIdx.x;  // 0..1023
    if (g >= 2 * HID) return;
    const float* W = (g < HID) ? W1 : W2;
    const int k = g & (HID - 1);
    float s = 0.f;
    for (int j = 0; j < HID; ++j) s += W[(size_t)j * HID + k] * v[j];
    ((g < HID) ? u1 : u2)[k] = s;
}

// ---------------------------------------------------------------------------
// a[b,n] = dot(src_out[b,n,:], u1);  d[b,r] = dot(tgt_out[b,r,:], u2)
// ---------------------------------------------------------------------------
__global__ void __launch_bounds__(256) ad_kernel(
    const float* __restrict__ src_out, const float* __restrict__ tgt_out,
    const float* __restrict__ u1, const float* __restrict__ u2,
    float* __restrict__ a, float* __restrict__ d) {
    const int lane = threadIdx.x & 31;
    const int row  = blockIdx.x * (blockDim.x >> 5) + (threadIdx.x >> 5);
    const int total = BATCH * NSEQ;
    const float* base;
    const float* u;
    float* o;
    int r;
    if (row < total)       { r = row;         base = src_out + (size_t)r * HID; u = u1; o = a; }
    else if (row < 2*total){ r = row - total; base = tgt_out + (size_t)r * HID; u = u2; o = d; }
    else return;
    float s = 0.f;
#pragma unroll
    for (int k = 0; k < HID / 32; ++k) s += base[k * 32 + lane] * u[k * 32 + lane];
#pragma unroll
    for (int off = 16; off; off >>= 1) s += __shfl_xor(s, off, 32);
    if (lane == 0) o[r] = s;
}

// ---------------------------------------------------------------------------
// Greedy pointer decode: one wave per batch, 8 columns per lane, sequential
// over rows.  Masked softmax + first-max argmax + mask removal.
// ---------------------------------------------------------------------------
__global__ void __launch_bounds__(256) pointer_kernel(
    const float* __restrict__ a,   // [B, N]
    const float* __restrict__ d,   // [B, N]
    const int*   __restrict__ ns_src,
    const int*   __restrict__ ns_tgt,
    float* __restrict__ out)       // [B, N, N]
{
    const int lane = threadIdx.x & 31;
    const int b = blockIdx.x * (blockDim.x >> 5) + (threadIdx.x >> 5);
    if (b >= BATCH) return;
    const int ls = ns_src[b];
    const int lt = ns_tgt[b];

    float av[8];
    unsigned mbits = 0;
#pragma unroll
    for (int k = 0; k < 8; ++k) {
        const int col = k * 32 + lane;
        av[k] = a[(size_t)b * NSEQ + col];
        if (col < lt) mbits |= (1u << k);
    }
    float* orow = out + (size_t)b * NSEQ * NSEQ;
    const float NEG = -3.4028234663852886e38f;

    for (int r = 0; r < NSEQ; ++r) {
        const float dr = d[(size_t)b * NSEQ + r];
        const bool active = r < ls;

        float u[8];
        float mx = NEG;
#pragma unroll
        for (int k = 0; k < 8; ++k) {
            u[k] = av[k] + dr;  // ALPHA == 1
            if ((mbits >> k) & 1u) mx = fmaxf(mx, u[k]);
        }
#pragma unroll
        for (int off = 16; off; off >>= 1) mx = fmaxf(mx, __shfl_xor(mx, off, 32));

        float e[8];
        float s = 0.f;
#pragma unroll
        for (int k = 0; k < 8; ++k) {
            e[k] = ((mbits >> k) & 1u) ? expf(u[k] - mx) : 0.f;
            s += e[k];
        }
#pragma unroll
        for (int off = 16; off; off >>= 1) s += __shfl_xor(s, off, 32);
        const float inv = (s > 0.f) ? 1.f / s : 0.f;

        float bv = -1.f;
        int   bc = NSEQ;
#pragma unroll
        for (int k = 0; k < 8; ++k) {
            const int col = k * 32 + lane;
            const bool m = (mbits >> k) & 1u;
            const float row = (m && active) ? e[k] * inv : 0.f;
            orow[(size_t)r * NSEQ + col] = row;
            const float cand = m ? row : -1.f;  // argmax(where(mask,row,-1))
            if (cand > bv || (cand == bv && col < bc)) { bv = cand; bc = col; }
        }
#pragma unroll
        for (int off = 16; off; off >>= 1) {
            const float ov = __shfl_xor(bv, off, 32);
            const int   oc = __shfl_xor(bc, off, 32);
            if (ov > bv || (ov == bv && oc < bc)) { bv = ov; bc = oc; }
        }
        if (active && bv > 0.f && (bc & 31) == lane) mbits &= ~(1u << (bc >> 5));
    }
}

// ---------------------------------------------------------------------------
// Host-side orchestration
// ---------------------------------------------------------------------------
extern "C" void kernel_launch(void* const* d_in, const int* in_sizes, int n_in,
                              void* d_out, int out_size, void* d_ws, size_t ws_size,
                              hipStream_t stream) {
    (void)in_sizes; (void)n_in; (void)out_size; (void)ws_size;

    const float* seq_src = (const float*)d_in[0];
    const float* seq_tgt = (const float*)d_in[1];
    const int*   ns_src  = (const int*)d_in[2];
    const int*   ns_tgt  = (const int*)d_in[3];
    const float* enc_Wih = (const float*)d_in[4];
    const float* enc_Whh = (const float*)d_in[5];
    const float* enc_bih = (const float*)d_in[6];
    const float* enc_bhh = (const float*)d_in[7];
    const float* dec_Wih = (const float*)d_in[8];
    const float* dec_Whh = (const float*)d_in[9];
    const float* dec_bih = (const float*)d_in[10];
    const float* dec_bhh = (const float*)d_in[11];
    const float* att_v   = (const float*)d_in[12];
    const float* att_W1  = (const float*)d_in[13];
    const float* att_W2  = (const float*)d_in[14];

    float* ws = (float*)d_ws;
    size_t off = 0;
    float* xprojT  = ws + off; off += (size_t)NSEQ * BATCH * GATESZ;  // 33.55M
    float* src_out = ws + off; off += (size_t)BATCH * NSEQ * HID;     //  8.39M
    float* tgt_out = ws + off; off += (size_t)BATCH * NSEQ * HID;     //  8.39M
    float* hbuf0   = ws + off; off += (size_t)BATCH * HID;
    float* hbuf1   = ws + off; off += (size_t)BATCH * HID;
    float* c       = ws + off; off += (size_t)BATCH * HID;
    float* u1      = ws + off; off += HID;
    float* u2      = ws + off; off += HID;
    float* a       = ws + off; off += (size_t)BATCH * NSEQ;
    float* d       = ws + off; off += (size_t)BATCH * NSEQ;
    unsigned* bar  = (unsigned*)(ws + off); off += 16;

    // barrier + encoder h0/c0 = 0  (encoder reads hbuf0 at t=0)
    init_kernel<<<1, 256, 0, stream>>>(bar, hbuf0, c);
    // attention vector projections (independent)
    uvec_kernel<<<4, 256, 0, stream>>>(att_W1, att_W2, att_v, u1, u2);

    // encoder: input projection then fused recurrence
    xproj_kernel<<<4096, 256, 0, stream>>>(seq_src, enc_Wih, enc_bih, enc_bhh, xprojT);
    lstm_kernel<<<16, 256, 0, stream>>>(xprojT, enc_Whh, ns_src, hbuf0, hbuf1, c,
                                        src_out, bar);
    // decoder: reuse xproj buffer; NSEQ even -> encoder final h is in hbuf0,
    // decoder t=0 reads hbuf0, so state carries over with the same pointers.
    xproj_kernel<<<4096, 256, 0, stream>>>(seq_tgt, dec_Wih, dec_bih, dec_bhh, xprojT);
    lstm_kernel<<<16, 256, 0, stream>>>(xprojT, dec_Whh, ns_tgt, hbuf0, hbuf1, c,
                                        tgt_out, bar);

    // attention scalars and greedy decode
    ad_kernel<<<4096, 256, 0, stream>>>(src_out, tgt_out, u1, u2, a, d);
    pointer_kernel<<<8, 256, 0, stream>>>(a, d, ns_src, ns_tgt, (float*)d_out);
}